// SAGE_43396349559222
// MI455X (gfx1250) — compile-verified
//
#include <hip/hip_runtime.h>

typedef __attribute__((ext_vector_type(2))) float v2f;
typedef __attribute__((ext_vector_type(8))) float v8f;
typedef __attribute__((ext_vector_type(4))) unsigned int v4u;
typedef __attribute__((ext_vector_type(4))) int v4i;
typedef __attribute__((ext_vector_type(8))) int v8i;

#define N_NODES   100000
#define N_EDGES   800000
#define IN_FEATS  100
#define N_HIDDEN  64
#define N_CLASSES 47

// ---------------------------------------------------------------------------
// Tensor Data Mover: 1-D contiguous global->LDS DMA (CDNA5 ISA ch.8 D# layout)
// Each GEMM block's A-tile is 16 consecutive rows x full width = one
// contiguous chunk, so a 1-D descriptor suffices:
//   group0: count=1 | lds_addr | global_addr(57b) | type=2
//   group1: data_size=4B, tensor_dim0=tile_dim0=n_elems, tensor_dim1=1,
//           tile_dim1=1, stride0=n_elems, no padding/multicast/iterate
// ---------------------------------------------------------------------------
#if defined(__HIP_DEVICE_COMPILE__) && __has_builtin(__builtin_amdgcn_tensor_load_to_lds)
#define SAGE_HAS_TDM 1
#else
#define SAGE_HAS_TDM 0
#endif

#if SAGE_HAS_TDM
__device__ __forceinline__ void tdm_load_1d(unsigned lds_off,
                                            const float* gptr, int n_elems) {
  unsigned long long ga = (unsigned long long)(size_t)gptr;
  v4u g0;
  g0[0] = 1u;                                   // count=1, user mode
  g0[1] = lds_off;                              // lds_addr (bytes)
  g0[2] = (unsigned)(ga & 0xffffffffu);         // global_addr[31:0]
  g0[3] = (unsigned)((ga >> 32) & 0x01ffffffu)  // global_addr[56:32]
          | (2u << 30);                         // type=2 ("image")
  unsigned ne = (unsigned)n_elems;
  v8i g1;
  g1[0] = (int)(2u << 16);                      // data_size = 4 bytes
  g1[1] = (int)((ne & 0xffffu) << 16);          // tensor_dim0[15:0]
  g1[2] = (int)((ne >> 16) | (1u << 16));       // tensor_dim0[31:16] | tensor_dim1=1
  g1[3] = (int)((ne & 0xffffu) << 16);          // tensor_dim1 hi=0 | tile_dim0
  g1[4] = 1;                                    // tile_dim1=1, tile_dim2=0
  g1[5] = (int)ne;                              // tensor_dim0_stride[31:0]
  g1[6] = 0;                                    // stride0 hi | stride1 lo
  g1[7] = 0;                                    // stride1 hi
  v4i gz = {0, 0, 0, 0};
#if __clang_major__ >= 23
  v8i gz8 = {0, 0, 0, 0, 0, 0, 0, 0};
  __builtin_amdgcn_tensor_load_to_lds(g0, g1, gz, gz, gz8, 0);
#else
  __builtin_amdgcn_tensor_load_to_lds(g0, g1, gz, gz, 0);
#endif
}
#endif

// ---------------------------------------------------------------------------
// Zero-fill with b128 stores (all our sizes are multiples of 4 floats)
// ---------------------------------------------------------------------------
__global__ void zero_kernel(float4* __restrict__ p, int n4) {
  int i = blockIdx.x * blockDim.x + threadIdx.x;
  if (i < n4) p[i] = make_float4(0.f, 0.f, 0.f, 0.f);
}

// ---------------------------------------------------------------------------
// In-degree: deg[dst[e]] += 1
// ---------------------------------------------------------------------------
__global__ void degree_kernel(const int* __restrict__ dst,
                              float* __restrict__ deg, int n_edges) {
  int e = blockIdx.x * blockDim.x + threadIdx.x;
  if (e < n_edges) atomicAdd(&deg[dst[e]], 1.0f);
}

// ---------------------------------------------------------------------------
// Edge aggregation: agg[dst[e], f:f+4] += h[src[e], f:f+4]
// One thread per (edge, float4 chunk): b128 gather + 4 f32 atomics; index
// arrays read once per 4 features; D is compile-time so the chunk math is
// strength-reduced.
// ---------------------------------------------------------------------------
template <int D>
__global__ void aggregate_kernel(const float* __restrict__ h,
                                 const int* __restrict__ src,
                                 const int* __restrict__ dst,
                                 float* __restrict__ agg, int n_edges) {
  constexpr int CH = D / 4;
  int i = blockIdx.x * blockDim.x + threadIdx.x;
  if (i >= n_edges * CH) return;
  int e = i / CH;
  int c = i - e * CH;
  const float4 v =
      ((const float4*)(h + (size_t)src[e] * D))[c];
  float* a = agg + (size_t)dst[e] * D + (size_t)c * 4;
  atomicAdd(a + 0, v.x);
  atomicAdd(a + 1, v.y);
  atomicAdd(a + 2, v.z);
  atomicAdd(a + 3, v.w);
}

// ---------------------------------------------------------------------------
// Mean: agg[n, f] /= max(deg[n], 1)   (in place, b128)
// ---------------------------------------------------------------------------
template <int D>
__global__ void mean_kernel(float* __restrict__ agg,
                            const float* __restrict__ deg, int n_nodes) {
  constexpr int CH = D / 4;
  int i = blockIdx.x * blockDim.x + threadIdx.x;
  if (i >= n_nodes * CH) return;
  int node = i / CH;
  float dv = fmaxf(deg[node], 1.0f);
  float4* p = (float4*)(agg + (size_t)node * D) + (i - node * CH);
  float4 v = *p;
  v.x /= dv; v.y /= dv; v.z /= dv; v.w /= dv;
  *p = v;
}

// ---------------------------------------------------------------------------
// Fused SAGE GEMM:  out = act( Hs @ Ws + Hn @ Wn + b )
//
// Block = 128 threads = 4 wave32s, one 16-row node tile per block.
// Wave 0 DMAs the two 16 x D_IN A-tiles into LDS via the Tensor Data Mover
// (contiguous 1-D copy), waits TENSORcnt, block barrier releases all waves.
// Each wave owns one 16-column output tile; K loop = chained
// V_WMMA_F32_16X16X4_F32 (full f32, matching the reference).
//
// VGPR layouts (CDNA5 ISA 7.12.2):
//   A 16x4 f32 : lane&15 = M row, lanes>=16 hold K+2; v0=K, v1=K+1
//   B 4x16 f32 : lane&15 = N col, lanes>=16 hold K+2 rows; v0=K, v1=K+1
//   C/D 16x16  : VGPR r -> M=r (lanes 0-15) / M=r+8 (lanes 16-31), N=lane&15
// ---------------------------------------------------------------------------
template <int RELU, int D_IN>
__global__ __launch_bounds__(128) void sage_gemm_kernel(
    const float* __restrict__ Hs, const float* __restrict__ Hn,
    const float* __restrict__ Ws, const float* __restrict__ Wn,
    const float* __restrict__ bias, float* __restrict__ out, int d_out) {
  __shared__ float lds_s[16 * D_IN];
  __shared__ float lds_n[16 * D_IN];

  const int m0  = blockIdx.x * 16;
  const int tid = threadIdx.x;

#if SAGE_HAS_TDM
  if (tid < 32) {  // wave 0 issues both DMAs (TDM ignores EXEC; per-wave op)
    tdm_load_1d((unsigned)(size_t)&lds_s[0], Hs + (size_t)m0 * D_IN,
                16 * D_IN);
    tdm_load_1d((unsigned)(size_t)&lds_n[0], Hn + (size_t)m0 * D_IN,
                16 * D_IN);
    __builtin_amdgcn_s_wait_tensorcnt(0);
  }
  __syncthreads();
#else
  // Fallback: manual b128 staging
  for (int idx = tid; idx < 16 * D_IN / 4; idx += 128) {
    ((float4*)lds_s)[idx] = ((const float4*)(Hs + (size_t)m0 * D_IN))[idx];
    ((float4*)lds_n)[idx] = ((const float4*)(Hn + (size_t)m0 * D_IN))[idx];
  }
  __syncthreads();
#endif

  const int wv   = tid >> 5;       // wave id -> N tile
  const int lane = tid & 31;
  const int n0   = wv * 16;
  if (n0 >= d_out) return;         // wave-uniform: EXEC stays all-ones below

  const int arow = lane & 15;            // A: M row
  const int koff = (lane >> 4) << 1;     // lanes 16-31 hold K+2
  const int col  = lane & 15;            // B/D: N col
  const int c    = n0 + col;
  const bool cval = (c < d_out);         // d_out = 47 tail guard
  const int ccl  = cval ? c : 0;         // clamp to keep loads in-bounds

  v8f acc = {0.f, 0.f, 0.f, 0.f, 0.f, 0.f, 0.f, 0.f};

  for (int kb = 0; kb < D_IN; kb += 4) {
    const int k0 = kb + koff;
    v2f a_s, a_n, b_s, b_n;
    a_s.x = lds_s[arow * D_IN + k0];
    a_s.y = lds_s[arow * D_IN + k0 + 1];
    a_n.x = lds_n[arow * D_IN + k0];
    a_n.y = lds_n[arow * D_IN + k0 + 1];
    b_s.x = Ws[k0 * d_out + ccl];
    b_s.y = Ws[(k0 + 1) * d_out + ccl];
    b_n.x = Wn[k0 * d_out + ccl];
    b_n.y = Wn[(k0 + 1) * d_out + ccl];
    if (!cval) {                    // lane select, re-converged before WMMA
      b_s = (v2f){0.f, 0.f};
      b_n = (v2f){0.f, 0.f};
    }
    // D = A_neigh*B_neigh + (A_self*B_self + C)  chained through accumulator
    acc = __builtin_amdgcn_wmma_f32_16x16x4_f32(
        false, a_s, false, b_s, (short)0, acc, false, false);
    acc = __builtin_amdgcn_wmma_f32_16x16x4_f32(
        false, a_n, false, b_n, (short)0, acc, false, false);
  }

  if (cval) {
    const float bb = bias[c];
    const int rbase = (lane < 16) ? 0 : 8;
    #pragma unroll
    for (int r = 0; r < 8; ++r) {
      float v = acc[r] + bb;
      if (RELU) v = fmaxf(v, 0.0f);
      out[(m0 + rbase + r) * d_out + c] = v;
    }
  }
}

// ---------------------------------------------------------------------------
// Host-side launch sequence (graph-capture safe: kernels only, all on stream)
// ---------------------------------------------------------------------------
extern "C" void kernel_launch(void* const* d_in, const int* in_sizes, int n_in,
                              void* d_out, int out_size, void* d_ws,
                              size_t ws_size, hipStream_t stream) {
  const float* x    = (const float*)d_in[0];
  const int*   esrc = (const int*)d_in[1];
  const int*   edst = (const int*)d_in[2];
  const float* Ws0  = (const float*)d_in[3];
  const float* Wn0  = (const float*)d_in[4];
  const float* b0   = (const float*)d_in[5];
  const float* Ws1  = (const float*)d_in[6];
  const float* Wn1  = (const float*)d_in[7];
  const float* b1   = (const float*)d_in[8];
  const float* Ws2  = (const float*)d_in[9];
  const float* Wn2  = (const float*)d_in[10];
  const float* b2   = (const float*)d_in[11];
  float* out = (float*)d_out;

  // Workspace layout (f32): deg[N] | agg[N*100] | h1[N*64] | h2[N*64]
  float* deg = (float*)d_ws;
  float* agg = deg + N_NODES;
  float* h1  = agg + (size_t)N_NODES * IN_FEATS;
  float* h2  = h1 + (size_t)N_NODES * N_HIDDEN;

  auto cdiv = [](int a, int b) { return (a + b - 1) / b; };
  const int gemm_blocks = N_NODES / 16;  // 100000 % 16 == 0

  // In-degrees (reused by all three layers).
  zero_kernel<<<cdiv(N_NODES / 4, 256), 256, 0, stream>>>(
      (float4*)deg, N_NODES / 4);
  degree_kernel<<<cdiv(N_EDGES, 256), 256, 0, stream>>>(edst, deg, N_EDGES);

  // ---- Layer 0: [N,100] -> [N,64], ReLU ----
  zero_kernel<<<cdiv(N_NODES * IN_FEATS / 4, 256), 256, 0, stream>>>(
      (float4*)agg, N_NODES * IN_FEATS / 4);
  aggregate_kernel<IN_FEATS>
      <<<cdiv(N_EDGES * (IN_FEATS / 4), 256), 256, 0, stream>>>(
          x, esrc, edst, agg, N_EDGES);
  mean_kernel<IN_FEATS>
      <<<cdiv(N_NODES * (IN_FEATS / 4), 256), 256, 0, stream>>>(
          agg, deg, N_NODES);
  sage_gemm_kernel<1, IN_FEATS><<<gemm_blocks, 128, 0, stream>>>(
      x, agg, Ws0, Wn0, b0, h1, N_HIDDEN);

  // ---- Layer 1: [N,64] -> [N,64], ReLU ----
  zero_kernel<<<cdiv(N_NODES * N_HIDDEN / 4, 256), 256, 0, stream>>>(
      (float4*)agg, N_NODES * N_HIDDEN / 4);
  aggregate_kernel<N_HIDDEN>
      <<<cdiv(N_EDGES * (N_HIDDEN / 4), 256), 256, 0, stream>>>(
          h1, esrc, edst, agg, N_EDGES);
  mean_kernel<N_HIDDEN>
      <<<cdiv(N_NODES * (N_HIDDEN / 4), 256), 256, 0, stream>>>(
          agg, deg, N_NODES);
  sage_gemm_kernel<1, N_HIDDEN><<<gemm_blocks, 128, 0, stream>>>(
      h1, agg, Ws1, Wn1, b1, h2, N_HIDDEN);

  // ---- Layer 2: [N,64] -> [N,47], no activation ----
  zero_kernel<<<cdiv(N_NODES * N_HIDDEN / 4, 256), 256, 0, stream>>>(
      (float4*)agg, N_NODES * N_HIDDEN / 4);
  aggregate_kernel<N_HIDDEN>
      <<<cdiv(N_EDGES * (N_HIDDEN / 4), 256), 256, 0, stream>>>(
          h2, esrc, edst, agg, N_EDGES);
  mean_kernel<N_HIDDEN>
      <<<cdiv(N_NODES * (N_HIDDEN / 4), 256), 256, 0, stream>>>(
          agg, deg, N_NODES);
  sage_gemm_kernel<0, N_HIDDEN><<<gemm_blocks, 128, 0, stream>>>(
      h2, agg, Ws2, Wn2, b2, out, N_CLASSES);
}